// DurationAdaptor_44882408243323
// MI455X (gfx1250) — compile-verified
//
#include <hip/hip_runtime.h>
#include <math.h>

typedef __attribute__((ext_vector_type(2))) float v2f;
typedef __attribute__((ext_vector_type(8))) float v8f;

#define B_   16
#define T_EN 1024
#define C_IN 512
#define C_H  256
#define KW   5
#define PADW 2

// ---------------------------------------------------------------------------
// Weight repack, K-pair-major: wTp[((kk/2)*C_H + co)*2 + (kk&1)] = B[kk, co]
// where B[kk, co] = w[co, ci, k], kk = k*CIN + ci.
// A lane's WMMA B-fragment (rows kk, kk+1 at column co) is then one aligned
// 8-byte load, 128B-contiguous across lanes 0-15.
// ---------------------------------------------------------------------------
__global__ void wt_kernel(const float* __restrict__ w, float* __restrict__ wT, int CIN) {
    int idx = blockIdx.x * 256 + threadIdx.x;
    int total = C_H * CIN * KW;
    if (idx >= total) return;
    int j  = idx & 1;                 // kk parity
    int co = (idx >> 1) & (C_H - 1);  // column
    int p  = idx >> 9;                // kk pair
    int kk = 2 * p + j;
    int ci = kk % CIN;
    int k  = kk / CIN;
    wT[idx] = w[((size_t)co * CIN + ci) * KW + k];
}

// ---------------------------------------------------------------------------
// Fused Conv1d(K=5, pad=2) + bias + ReLU + LayerNorm(C_H) via f32 WMMA.
// One workgroup: 16 time rows x all 256 output channels (8 waves x 2 tiles).
// A-operand (shared by all 8 waves) staged in LDS chunk-wise (64 K-cols, one
// conv tap per chunk); boundary check amortized to once per chunk.
// mode 0: write normalized h [B,T,C_H]
// mode 1: dot with lin_w + lin_b + duration mask -> log_duration_pred [B,T]
// ---------------------------------------------------------------------------
template <int CIN>
__global__ __launch_bounds__(256) void conv_ln_kernel(
    const float* __restrict__ x,     // [B,T,CIN]
    const float* __restrict__ wT,    // K-pair-major packed weights
    const float* __restrict__ bias,  // [C_H]
    const float* __restrict__ gma,   // [C_H]
    const float* __restrict__ bta,   // [C_H]
    const float* __restrict__ lw,    // [C_H]   (mode 1)
    const float* __restrict__ lb,    // [1]     (mode 1)
    const float* __restrict__ dtar,  // [B,T]   (mask source, mode 1)
    float* __restrict__ out,         // mode0: [B,T,C_H]; mode1: [B,T]
    int mode) {
    constexpr int KK = KW * CIN;
    constexpr int KC = 64;           // K-chunk (stays within one conv tap)
    constexpr int AS = KC + 4;       // LDS row stride 68 -> conflict-free b64 frag reads

    __shared__ float As[16][AS];
    __shared__ float h[16][264];
    __shared__ float red[16][16];
    __shared__ float red2[16][16];
    __shared__ float mean_s[16];
    __shared__ float rstd_s[16];

    const int tid  = threadIdx.x;
    const int wave = tid >> 5;
    const int lane = tid & 31;
    const int m    = lane & 15;      // A row
    const int kh   = lane >> 4;      // K-half (lanes 16-31 hold K=2,3)

    const int gtile = blockIdx.x;    // over B*T/16; tiles never cross batch
    const int b     = gtile >> 6;    // 1024/16 = 64 tiles per batch
    const int t0    = (gtile & 63) << 4;

    const int n0 = wave * 32 + (lane & 15);  // B/D column (tile 0)
    const int n1 = n0 + 16;                  // tile 1

    const float* xb = x + (size_t)b * T_EN * CIN;

    // staging coords (per thread): row r, 4 contiguous cols
    const int sr = tid >> 4;         // 0..15
    const int sc = (tid & 15) << 2;  // 0,4,...,60

    v8f acc0 = {0.f, 0.f, 0.f, 0.f, 0.f, 0.f, 0.f, 0.f};
    v8f acc1 = acc0;

    for (int c0 = 0; c0 < KK; c0 += KC) {
        const int k   = c0 / CIN;                 // power-of-two CIN -> shift
        const int ci0 = c0 - k * CIN;

        // ---- stage A chunk [16 x 64] into LDS (one boundary check per chunk)
        {
            const int t = t0 + sr + k - PADW;
            float4 v = make_float4(0.f, 0.f, 0.f, 0.f);
            if (t >= 0 && t < T_EN)
                v = *(const float4*)(xb + (size_t)t * CIN + (ci0 + sc));
            As[sr][sc + 0] = v.x;
            As[sr][sc + 1] = v.y;
            As[sr][sc + 2] = v.z;
            As[sr][sc + 3] = v.w;
        }
        __syncthreads();

        // ---- 16 WMMA K-steps over the chunk (fully unrolled)
#pragma unroll
        for (int q = 0; q < KC; q += 4) {
            v2f a;
            a.x = As[m][q + 2 * kh];
            a.y = As[m][q + 2 * kh + 1];
            // pair row = (c0+q)/2 + kh; fragment = one aligned b64 load
            const float* pb = wT + ((size_t)(((c0 + q) >> 1) + kh)) * (C_H * 2);
            v2f b0 = *(const v2f*)(pb + n0 * 2);
            v2f b1 = *(const v2f*)(pb + n1 * 2);
            acc0 = __builtin_amdgcn_wmma_f32_16x16x4_f32(false, a, false, b0, (short)0, acc0, false, false);
            acc1 = __builtin_amdgcn_wmma_f32_16x16x4_f32(false, a, false, b1, (short)0, acc1, false, false);
        }
        __syncthreads();
    }

    // ---- bias + ReLU -> LDS tile (D layout: VGPR r holds row r + 8*kh)
    const float bia0 = bias[n0];
    const float bia1 = bias[n1];
#pragma unroll
    for (int r = 0; r < 8; ++r) {
        int row = r + 8 * kh;
        h[row][n0] = fmaxf(acc0[r] + bia0, 0.f);
        h[row][n1] = fmaxf(acc1[r] + bia1, 0.f);
    }
    __syncthreads();

    // ---- LayerNorm stats: 16 threads per row
    const int row = tid >> 4;
    const int j   = tid & 15;
    float s = 0.f, s2 = 0.f;
#pragma unroll
    for (int i = 0; i < 16; ++i) {
        float v = h[row][j + 16 * i];
        s += v; s2 += v * v;
    }
    red[row][j] = s; red2[row][j] = s2;
    __syncthreads();
    if (j == 0) {
        float S = 0.f, S2 = 0.f;
#pragma unroll
        for (int q = 0; q < 16; ++q) { S += red[row][q]; S2 += red2[row][q]; }
        float mu  = S * (1.f / 256.f);
        float var = S2 * (1.f / 256.f) - mu * mu;
        mean_s[row] = mu;
        rstd_s[row] = rsqrtf(var + 1e-5f);
    }
    __syncthreads();

    const float mu = mean_s[row], rs = rstd_s[row];
    if (mode == 0) {
        float* o = out + ((size_t)b * T_EN + t0 + row) * C_H;
#pragma unroll
        for (int i = 0; i < 16; ++i) {
            int c = j + 16 * i;
            o[c] = (h[row][c] - mu) * rs * gma[c] + bta[c];
        }
    } else {
        float p = 0.f;
#pragma unroll
        for (int i = 0; i < 16; ++i) {
            int c = j + 16 * i;
            p += ((h[row][c] - mu) * rs * gma[c] + bta[c]) * lw[c];
        }
        __syncthreads();
        red[row][j] = p;
        __syncthreads();
        if (j == 0) {
            float S = 0.f;
#pragma unroll
            for (int q = 0; q < 16; ++q) S += red[row][q];
            size_t ridx = (size_t)b * T_EN + t0 + row;
            // src_mask True <=> duration_target == 0 -> masked to 0
            out[ridx] = (dtar[ridx] == 0.f) ? 0.f : (S + lb[0]);
        }
    }
}

// ---------------------------------------------------------------------------
// Per-batch prefix cumsum of duration_target and exp(ldp)-1 (leading 0).
// ---------------------------------------------------------------------------
__global__ void scan_kernel(const float* __restrict__ dt, const float* __restrict__ ldp,
                            float* __restrict__ cum_dt, float* __restrict__ cum_dp) {
    int b = blockIdx.x;
    if (threadIdx.x != 0) return;
    float a = 0.f, c = 0.f;
    cum_dt[b * (T_EN + 1)] = 0.f;
    cum_dp[b * (T_EN + 1)] = 0.f;
    for (int mm = 0; mm < T_EN; ++mm) {
        a += dt[b * T_EN + mm];
        c += expf(ldp[b * T_EN + mm]) - 1.f;
        cum_dt[b * (T_EN + 1) + mm + 1] = a;
        cum_dp[b * (T_EN + 1) + mm + 1] = c;
    }
}

// ---------------------------------------------------------------------------
// src_idx[b,n] = first m with (float)n < cum_dt[b,m+1]   (binary search)
// ---------------------------------------------------------------------------
__global__ void srcidx_kernel(const float* __restrict__ cum_dt, int* __restrict__ si, int TDE) {
    int idx = blockIdx.x * 256 + threadIdx.x;
    if (idx >= B_ * TDE) return;
    int b = idx / TDE;
    int n = idx - b * TDE;
    const float* c = cum_dt + b * (T_EN + 1);
    float fn = (float)n;
    int lo = 0, hi = T_EN;
    while (lo < hi) {
        int mid = (lo + hi) >> 1;
        if (fn < c[mid + 1]) hi = mid; else lo = mid + 1;
    }
    si[idx] = lo;
}

// ---------------------------------------------------------------------------
// Output 1: alignments_duration_pred [B, T_EN, TDE]
// ---------------------------------------------------------------------------
__global__ void align_dp_kernel(const float* __restrict__ cum_dp, const float* __restrict__ dtar,
                                const int* __restrict__ mel, float* __restrict__ out, int TDE) {
    int row = blockIdx.x;            // b*T_EN + m
    int b = row >> 10;
    int m = row & (T_EN - 1);
    const float* c = cum_dp + b * (T_EN + 1);
    float lo = c[m], hi = c[m + 1];
    float xm = (dtar[row] == 0.f) ? 1.f : 0.f;   // src_mask (True at padding)
    int ml = mel[b];
    float* o = out + (size_t)row * TDE;
    for (int n = threadIdx.x; n < TDE; n += 256) {
        float fn = (float)n;
        float v = (((fn < hi) ? 1.f : 0.f) - ((fn < lo) ? 1.f : 0.f)) * xm;
        o[n] = (n < ml) ? v : 0.f;
    }
}

// ---------------------------------------------------------------------------
// Output 4: attn^T [B, TDE, T_EN] — one-hot rows from src_idx
// ---------------------------------------------------------------------------
__global__ void attnT_kernel(const int* __restrict__ si, const int* __restrict__ mel,
                             float* __restrict__ out, int TDE) {
    int row = blockIdx.x;            // b*TDE + n
    int b = row / TDE;
    int n = row - b * TDE;
    int mi = (n < mel[b]) ? si[row] : -1;
    float* o = out + (size_t)row * T_EN;
    for (int m = threadIdx.x; m < T_EN; m += 256)
        o[m] = (m == mi) ? 1.f : 0.f;
}

// ---------------------------------------------------------------------------
// Output 3: encoder_output_dr [B, TDE, C_IN] — gather via src_idx
// ---------------------------------------------------------------------------
__global__ void gather_kernel(const float* __restrict__ enc, const int* __restrict__ si,
                              const int* __restrict__ mel, float* __restrict__ out, int TDE) {
    int row = blockIdx.x;            // b*TDE + n
    int b = row / TDE;
    int n = row - b * TDE;
    bool valid = (n < mel[b]);
    int mi = valid ? si[row] : 0;
    const float* src = enc + ((size_t)b * T_EN + mi) * C_IN;
    float* o = out + (size_t)row * C_IN;
    for (int c = threadIdx.x; c < C_IN; c += 256)
        o[c] = valid ? src[c] : 0.f;
}

// ---------------------------------------------------------------------------
extern "C" void kernel_launch(void* const* d_in, const int* in_sizes, int n_in,
                              void* d_out, int out_size, void* d_ws, size_t ws_size,
                              hipStream_t stream) {
    const float* enc  = (const float*)d_in[0];   // encoder_output
    const float* encr = (const float*)d_in[1];   // encoder_output_res
    const float* dtar = (const float*)d_in[2];   // duration_target
    const float* c1w  = (const float*)d_in[3];
    const float* c1b  = (const float*)d_in[4];
    const float* g1   = (const float*)d_in[5];
    const float* b1   = (const float*)d_in[6];
    const float* c2w  = (const float*)d_in[7];
    const float* c2b  = (const float*)d_in[8];
    const float* g2   = (const float*)d_in[9];
    const float* b2   = (const float*)d_in[10];
    const float* lw   = (const float*)d_in[11];
    const float* lb   = (const float*)d_in[12];
    const int*   mel  = (const int*)d_in[14];

    // out_size = B*T_EN*TDE + B*T_EN + B*TDE*C_IN + B*TDE*T_EN
    int TDE = (int)(((long long)out_size - (long long)B_ * T_EN) /
                    ((long long)2 * B_ * T_EN + (long long)B_ * C_IN));

    float* out1 = (float*)d_out;                      // [B,T_EN,TDE]
    float* out2 = out1 + (size_t)B_ * T_EN * TDE;     // [B,T_EN]
    float* out3 = out2 + (size_t)B_ * T_EN;           // [B,TDE,C_IN]
    float* out4 = out3 + (size_t)B_ * TDE * C_IN;     // [B,TDE,T_EN]

    float* ws     = (float*)d_ws;
    float* wT1    = ws;  ws += (size_t)KW * C_IN * C_H;
    float* wT2    = ws;  ws += (size_t)KW * C_H * C_H;
    float* h1     = ws;  ws += (size_t)B_ * T_EN * C_H;
    float* cum_dt = ws;  ws += (size_t)B_ * (T_EN + 1);
    float* cum_dp = ws;  ws += (size_t)B_ * (T_EN + 1);
    int*   si     = (int*)ws;                             // B*TDE ints

    wt_kernel<<<(C_H * C_IN * KW + 255) / 256, 256, 0, stream>>>(c1w, wT1, C_IN);
    wt_kernel<<<(C_H * C_H * KW + 255) / 256, 256, 0, stream>>>(c2w, wT2, C_H);

    conv_ln_kernel<C_IN><<<B_ * T_EN / 16, 256, 0, stream>>>(
        encr, wT1, c1b, g1, b1, lw, lb, dtar, h1, 0);
    conv_ln_kernel<C_H><<<B_ * T_EN / 16, 256, 0, stream>>>(
        h1, wT2, c2b, g2, b2, lw, lb, dtar, out2, 1);

    scan_kernel<<<B_, 32, 0, stream>>>(dtar, out2, cum_dt, cum_dp);
    srcidx_kernel<<<(B_ * TDE + 255) / 256, 256, 0, stream>>>(cum_dt, si, TDE);

    align_dp_kernel<<<B_ * T_EN, 256, 0, stream>>>(cum_dp, dtar, mel, out1, TDE);
    attnT_kernel<<<B_ * TDE, 256, 0, stream>>>(si, mel, out4, TDE);
    gather_kernel<<<B_ * TDE, 256, 0, stream>>>(enc, si, mel, out3, TDE);
}